// NetModular_67551245631647
// MI455X (gfx1250) — compile-verified
//
#include <hip/hip_runtime.h>
#include <math.h>

#define G_    2048
#define N_    64
#define E_    256
#define NHID_ 64
#define K_    32
#define P_    200000

typedef float v2f __attribute__((ext_vector_type(2)));
typedef float v8f __attribute__((ext_vector_type(8)));

// ---------------------------------------------------------------------------
// Kernel 1: per-graph GCN + SAGPool + global max/mean pool, fully in LDS.
// One 64-thread block (2 wave32) per graph; thread tid owns node tid.
// Exploits rank-1 structure (input features are all-ones): h[v] = relu(c[v]*W+b)
// so all scatters are scalar LDS atomics (ds_add_f32), not 64-vector ones.
// ---------------------------------------------------------------------------
__global__ __launch_bounds__(64) void graph_feat_kernel(
    const int*   __restrict__ graph_edges,   // [G,2,E]
    const float* __restrict__ conv1_w,       // [1,64]
    const float* __restrict__ conv1_b,       // [64]
    const float* __restrict__ w_rel,         // [64,1]
    const float* __restrict__ b_rel,         // [1]
    const float* __restrict__ w_root,        // [64,1]
    float*       __restrict__ mf)            // [G,128] out
{
  __shared__ int   srcE[E_], dstE[E_];
  __shared__ float Wsh[NHID_], Bsh[NHID_], WRel[NHID_], WRoot[NHID_];
  __shared__ float deg[N_], dinv[N_], cval[N_], tval[N_], score[N_], gate[N_];
  __shared__ int   sel[N_];

  const int tid = threadIdx.x;
  const int g   = blockIdx.x;
  const int* eg = graph_edges + (size_t)g * 2 * E_;

  Wsh[tid]   = conv1_w[tid];
  Bsh[tid]   = conv1_b[tid];
  WRel[tid]  = w_rel[tid];
  WRoot[tid] = w_root[tid];
  #pragma unroll
  for (int i = 0; i < 4; ++i) {
    const int e = tid + 64 * i;
    srcE[e] = eg[e];
    dstE[e] = eg[E_ + e];
  }
  deg[tid] = 1.0f;                      // self-loop weight
  __syncthreads();

  #pragma unroll
  for (int i = 0; i < 4; ++i)
    atomicAdd(&deg[dstE[tid + 64 * i]], 1.0f);
  __syncthreads();

  dinv[tid] = 1.0f / sqrtf(deg[tid]);   // deg^-0.5
  cval[tid] = 1.0f / deg[tid];          // self-loop term
  __syncthreads();

  #pragma unroll
  for (int i = 0; i < 4; ++i) {
    const int e = tid + 64 * i;
    atomicAdd(&cval[dstE[e]], dinv[srcE[e]] * dinv[dstE[e]]);
  }
  __syncthreads();

  // t[v] = h[v].w_rel ; score[v] init = h[v].w_root + b_rel
  {
    const float cv = cval[tid];
    float t = 0.f, r = 0.f;
    #pragma unroll 8
    for (int k = 0; k < NHID_; ++k) {
      const float hk = fmaxf(cv * Wsh[k] + Bsh[k], 0.f);
      t += hk * WRel[k];
      r += hk * WRoot[k];
    }
    tval[tid]  = t;
    score[tid] = r + b_rel[0];
  }
  __syncthreads();

  #pragma unroll
  for (int i = 0; i < 4; ++i) {
    const int e = tid + 64 * i;
    atomicAdd(&score[dstE[e]], tval[srcE[e]]);
  }
  __syncthreads();

  // top-k (K=32) via rank, jax.lax.top_k tie-break (lower index wins)
  {
    const float sv = score[tid];
    int rank = 0;
    #pragma unroll 8
    for (int u = 0; u < N_; ++u) {
      const float su = score[u];
      rank += (su > sv) || (su == sv && u < tid);
    }
    sel[tid]  = (rank < K_) ? 1 : 0;
    gate[tid] = tanhf(sv);
  }
  __syncthreads();

  // fused global max-pool || mean-pool over selected nodes; thread tid = channel
  {
    const float wk = Wsh[tid], bk = Bsh[tid];
    float mx = -INFINITY, sm = 0.f;
    #pragma unroll 8
    for (int v = 0; v < N_; ++v) {
      if (sel[v]) {
        const float val = fmaxf(cval[v] * wk + bk, 0.f) * gate[v];
        mx = fmaxf(mx, val);
        sm += val;
      }
    }
    float* o = mf + (size_t)g * 128;
    o[tid]      = mx;
    o[64 + tid] = sm * (1.0f / (float)K_);
  }
}

// ---------------------------------------------------------------------------
// Kernel 2: a[g] = mf[g].pred_w[0:128] + pred_b ; b[g] = mf[g].pred_w[128:256]
// as a 2048x128x2 f32 GEMM using V_WMMA_F32_16X16X4_F32 (N padded to 16,
// cols 0/1 carry pred_w halves, cols 2..15 are zero).
// One wave32 per 16-row tile; EXEC all-ones, no divergence before WMMA.
// A 16x4 layout: lanes 0-15 -> K {0,1}; lanes 16-31 -> K {2,3} (per lane M=lane&15).
// ---------------------------------------------------------------------------
__global__ __launch_bounds__(32) void edge_logit_precompute(
    const float* __restrict__ mf,      // [G,128]
    const float* __restrict__ pred_w,  // [256,1]
    const float* __restrict__ pred_b,  // [1]
    float*       __restrict__ aArr,    // [G]
    float*       __restrict__ bArr)    // [G]
{
  const int lane = threadIdx.x;
  const int m0   = blockIdx.x * 16;
  const int col  = lane & 15;
  const int half = lane >> 4;
  const int rowA = m0 + col;           // A row owned by this lane
  const int kofs = half * 2;           // K sub-offset per half-wave
  const float c0 = (col == 0) ? 1.f : 0.f;
  const float c1 = (col == 1) ? 1.f : 0.f;

  const float* __restrict__ arow = mf + (size_t)rowA * 128;

  v8f acc = {};
  #pragma unroll
  for (int kk = 0; kk < 32; ++kk) {
    const int k = kk * 4 + kofs;
    v2f a, b;
    a.x = arow[k];
    a.y = arow[k + 1];
    b.x = c0 * pred_w[k]     + c1 * pred_w[128 + k];
    b.y = c0 * pred_w[k + 1] + c1 * pred_w[128 + k + 1];
    acc = __builtin_amdgcn_wmma_f32_16x16x4_f32(
        /*neg_a=*/false, a, /*neg_b=*/false, b,
        /*c_mod=*/(short)0, acc, /*reuse_a=*/false, /*reuse_b=*/false);
  }

  // C/D layout: VGPR j holds rows (j, j+8); lanes 0-15 = cols 0-15 of rows 0-7,
  // lanes 16-31 = cols 0-15 of rows 8-15.
  const float pb = pred_b[0];
  if (col == 0) {
    #pragma unroll
    for (int j = 0; j < 8; ++j)
      aArr[m0 + half * 8 + j] = acc[j] + pb;
  } else if (col == 1) {
    #pragma unroll
    for (int j = 0; j < 8; ++j)
      bArr[m0 + half * 8 + j] = acc[j];
  }
}

// ---------------------------------------------------------------------------
// Kernel 3: out[p] = sigmoid(a[s[p]] + b[d[p]]); a/b tables (16KB) sit in L2.
// ---------------------------------------------------------------------------
__global__ __launch_bounds__(256) void ddi_sigmoid_kernel(
    const int*   __restrict__ ddi,     // [2,P]
    const float* __restrict__ aArr,
    const float* __restrict__ bArr,
    float*       __restrict__ out)     // [P]
{
  const int p = blockIdx.x * blockDim.x + threadIdx.x;
  if (p >= P_) return;
  const int s = ddi[p];
  const int d = ddi[P_ + p];
  const float z = aArr[s] + bArr[d];
  out[p] = 1.0f / (1.0f + expf(-z));
}

extern "C" void kernel_launch(void* const* d_in, const int* in_sizes, int n_in,
                              void* d_out, int out_size, void* d_ws, size_t ws_size,
                              hipStream_t stream) {
  const int*   graph_edges = (const int*)  d_in[0];
  const int*   ddi         = (const int*)  d_in[1];
  const float* conv1_w     = (const float*)d_in[2];
  const float* conv1_b     = (const float*)d_in[3];
  const float* w_rel       = (const float*)d_in[4];
  const float* b_rel       = (const float*)d_in[5];
  const float* w_root      = (const float*)d_in[6];
  const float* pred_w      = (const float*)d_in[7];
  const float* pred_b      = (const float*)d_in[8];
  float* out = (float*)d_out;

  float* mf   = (float*)d_ws;          // [G,128]  = 1 MB
  float* aArr = mf + (size_t)G_ * 128; // [G]
  float* bArr = aArr + G_;             // [G]

  graph_feat_kernel<<<G_, 64, 0, stream>>>(graph_edges, conv1_w, conv1_b,
                                           w_rel, b_rel, w_root, mf);
  edge_logit_precompute<<<G_ / 16, 32, 0, stream>>>(mf, pred_w, pred_b, aArr, bArr);
  ddi_sigmoid_kernel<<<(P_ + 255) / 256, 256, 0, stream>>>(ddi, aArr, bArr, out);
}